// EMAQuantizer_90967407329335
// MI455X (gfx1250) — compile-verified
//
#include <hip/hip_runtime.h>

typedef float v2f __attribute__((ext_vector_type(2)));
typedef float v8f __attribute__((ext_vector_type(8)));

constexpr int Cc   = 64;
constexpr int DHW  = 16 * 32 * 32;   // 16384
constexpr int NP   = 4 * DHW;        // 65536 points
constexpr int KC   = 1024;           // codebook entries
constexpr int TK   = 128;            // codes per LDS tile
constexpr int STR  = 2 * TK + 32;    // 288 floats: pair-row stride; 288%64==32 -> halves hit disjoint banks
constexpr int NBLK = KC / TK;        // 8
constexpr long CBOFF = (long)NP * Cc;    // 4194304 : codebook_loss
constexpr long CMOFF = CBOFF + 1;        // commitment_loss
constexpr long IXOFF = CBOFF + 2;        // indices (written as float)

// ---- Kernel 1: ||e_k||^2 into d_ws, zero the loss slots ----
__global__ __launch_bounds__(256) void vq_prep(const float* __restrict__ emb,
                                               float* __restrict__ enorm,
                                               float* __restrict__ out) {
    int k = blockIdx.x * 256 + threadIdx.x;
    if (k < KC) {
        const float* row = emb + (long)k * Cc;
        float s = 0.f;
        #pragma unroll
        for (int c = 0; c < Cc; ++c) { float v = row[c]; s = fmaf(v, v, s); }
        enorm[k] = s;
    }
    if (k < 2) out[CBOFF + k] = 0.f;
}

// ---- Kernel 2: WMMA distance matmul + argmin + quantize + losses ----
// Two 16-point groups per wave; embedding tiles stream through LDS via
// double-buffered GLOBAL_LOAD_ASYNC_TO_LDS_B64 (ASYNCcnt pipeline).
__global__ __launch_bounds__(256) void vq_main(const float* __restrict__ ze,
                                               const float* __restrict__ emb,
                                               const float* __restrict__ enorm,
                                               float* __restrict__ out) {
    __shared__ float sE[2][32 * STR];  // 2 x 36 KB: [channel-pair][code*2 + lo/hi]
    __shared__ float sN[KC];           // 4 KB: ||e_k||^2

    const int t    = threadIdx.x;
    const int lane = t & 31;
    const int wv   = t >> 5;
    const int p    = lane & 15;        // point col (B/D) and code row (A) index
    const int hi   = lane >> 4;

    for (int i = t; i < KC; i += 256) sN[i] = enorm[i];

    // Async tile fetch: element i of a tile is b64 # i (code k=i>>5, pair cp=i&31);
    // global source is contiguous (8*i bytes), LDS dest scatters the transpose.
    auto issue_tile = [&](int blk, int buf) {
        const unsigned long long src =
            (unsigned long long)(emb + (long)blk * TK * Cc);
        const unsigned ldsbase = (unsigned)(uintptr_t)&sE[buf][0];
        #pragma unroll
        for (int j = 0; j < 16; ++j) {
            int i  = t + 256 * j;
            int k  = i >> 5;
            int cp = i & 31;
            unsigned lds = ldsbase + (unsigned)(cp * (STR * 4) + k * 8);
            unsigned go  = (unsigned)(i * 8);
            asm volatile("global_load_async_to_lds_b64 %0, %1, %2"
                         :: "v"(lds), "v"(go), "s"(src) : "memory");
        }
    };

    const int g0 = (blockIdx.x * 8 + wv) * 2;  // first of two 16-point groups
    const int n0 = g0 * 16;                    // 32 consecutive points, same batch b
    const int b  = n0 / DHW;
    const int sp = (n0 % DHW) + p;
    const float* xb0 = ze + (long)b * Cc * DHW + sp;
    const float* xb1 = xb0 + 16;

    issue_tile(0, 0);                          // prologue: tile 0 in flight

    // B operands (4x16 K x N): lane<16 carries even channels of its point,
    // lane>=16 odd channels; v2f keeps each K-step's pair register-aligned.
    v2f pa[16], pb[16];
    #pragma unroll
    for (int s = 0; s < 16; ++s) {
        pa[s].x = xb0[(long)(4 * s + hi)     * DHW];
        pa[s].y = xb0[(long)(4 * s + hi + 2) * DHW];
        pb[s].x = xb1[(long)(4 * s + hi)     * DHW];
        pb[s].y = xb1[(long)(4 * s + hi + 2) * DHW];
    }

    float best0 = 3.4e38f, best1 = 3.4e38f;
    int   idx0  = 0,       idx1  = 0;

    for (int blk = 0; blk < NBLK; ++blk) {
        const int buf = blk & 1;
        asm volatile("s_wait_asynccnt 0x0" ::: "memory");  // my tile writes done
        __syncthreads();                                   // everyone's writes done,
                                                           // and prev compute on buf^1 done
        if (blk + 1 < NBLK) issue_tile(blk + 1, buf ^ 1);  // prefetch next tile

        const float* sEb = sE[buf];
        for (int ch = 0; ch < TK / 16; ++ch) {
            const int cb = ch * 16;
            v8f acc0 = {0.f,0.f,0.f,0.f,0.f,0.f,0.f,0.f};
            v8f acc1 = {0.f,0.f,0.f,0.f,0.f,0.f,0.f,0.f};
            #pragma unroll
            for (int s = 0; s < 16; ++s) {
                // A (16 codes x 4 channels): one aligned ds_load_b64 per step
                v2f a = *(const v2f*)&sEb[(2 * s + hi) * STR + 2 * (cb + p)];
                acc0 = __builtin_amdgcn_wmma_f32_16x16x4_f32(
                    false, a, false, pa[s], (short)0, acc0, false, false);
                acc1 = __builtin_amdgcn_wmma_f32_16x16x4_f32(
                    false, a, false, pb[s], (short)0, acc1, false, false);
            }
            const int mb = blk * TK + cb + 8 * hi;   // D: VGPR r -> code mb+r
            #pragma unroll
            for (int r = 0; r < 8; ++r) {
                float nrm = sN[mb + r];
                float c0 = fmaf(-2.f, acc0[r], nrm);
                float c1 = fmaf(-2.f, acc1[r], nrm);
                if (c0 < best0) { best0 = c0; idx0 = mb + r; }
                if (c1 < best1) { best1 = c1; idx1 = mb + r; }
            }
        }
    }

    // combine the two lane halves (disjoint code sets); symmetric result.
    {
        float ob = __shfl_xor(best0, 16, 32);
        int   oi = __shfl_xor(idx0, 16, 32);
        if (ob < best0 || (ob == best0 && oi < idx0)) { best0 = ob; idx0 = oi; }
        ob = __shfl_xor(best1, 16, 32);
        oi = __shfl_xor(idx1, 16, 32);
        if (ob < best1 || (ob == best1 && oi < idx1)) { best1 = ob; idx1 = oi; }
    }

    if (lane < 16) {
        out[IXOFF + n0 + p]      = (float)idx0;
        out[IXOFF + n0 + 16 + p] = (float)idx1;
    }

    // quantized_st == quantized numerically; gather code rows (hot in L2),
    // write output, accumulate squared error for both groups.
    const float* e0 = emb + (long)idx0 * Cc;
    const float* e1 = emb + (long)idx1 * Cc;
    float* q0 = out + (long)b * Cc * DHW + sp;
    float* q1 = q0 + 16;
    float lsum = 0.f;
    #pragma unroll
    for (int s = 0; s < 16; ++s) {
        int c0 = 4 * s + hi, c1 = c0 + 2;
        float a0 = e0[c0], a1 = e0[c1];
        float b0 = e1[c0], b1 = e1[c1];
        q0[(long)c0 * DHW] = a0;
        q0[(long)c1 * DHW] = a1;
        q1[(long)c0 * DHW] = b0;
        q1[(long)c1 * DHW] = b1;
        float d0 = a0 - pa[s].x, d1 = a1 - pa[s].y;
        float d2 = b0 - pb[s].x, d3 = b1 - pb[s].y;
        lsum = fmaf(d0, d0, fmaf(d1, d1, fmaf(d2, d2, fmaf(d3, d3, lsum))));
    }
    #pragma unroll
    for (int m = 1; m < 32; m <<= 1) lsum += __shfl_xor(lsum, m, 32);
    if (lane == 0) {
        float contrib = lsum * (1.0f / ((float)NP * (float)Cc));
        atomicAdd(out + CBOFF, contrib);   // codebook_loss
        atomicAdd(out + CMOFF, contrib);   // commitment_loss (numerically equal)
    }
}

extern "C" void kernel_launch(void* const* d_in, const int* in_sizes, int n_in,
                              void* d_out, int out_size, void* d_ws, size_t ws_size,
                              hipStream_t stream) {
    const float* ze  = (const float*)d_in[0];   // (4,64,16,32,32) f32
    const float* emb = (const float*)d_in[1];   // (1024,64) f32
    float* out   = (float*)d_out;
    float* enorm = (float*)d_ws;                // 1024 floats of scratch

    vq_prep<<<4, 256, 0, stream>>>(emb, enorm, out);
    vq_main<<<NP / (32 * 8), 256, 0, stream>>>(ze, emb, enorm, out);
}